// MissingSensorImputation_58909771432758
// MI455X (gfx1250) — compile-verified
//
#include <hip/hip_runtime.h>
#include <hip/hip_bf16.h>
#include <stddef.h>

// MI455X / gfx1250, wave32. Phase 1 (edge scatter-add) is the bottleneck
// (~5.2 GB of gather + f32-atomic traffic, L2-resident since 2x82MB < 192MB L2).
// Phase 2 (MLP) is tiny (15.7 GFLOP) -> use exact-precision f32 WMMA
// (V_WMMA_F32_16X16X4_F32) so the matrix pipe does the GEMMs.

typedef __attribute__((ext_vector_type(2))) float v2f;
typedef __attribute__((ext_vector_type(8))) float v8f;

static constexpr int B = 16, N = 10000, H = 128, E = 320000;
static constexpr int TPB   = N / 16;          // 625 node-tiles per batch
static constexpr int TILES = B * TPB;         // 10000 tiles of 16 rows
static constexpr int WAVES = 2;               // waves per block
static constexpr int CSTR  = 260;             // ctx row stride (256 + 4 pad) -> bank-conflict-free
static constexpr int HSTR  = 132;             // hidden row stride (128 + 4 pad)
static constexpr int WSZ   = 16 * CSTR + 16 * HSTR;  // floats of LDS per wave (6272)

__global__ void zero_f4(float4* p, int n4) {
  int i = blockIdx.x * blockDim.x + threadIdx.x;
  if (i < n4) p[i] = make_float4(0.f, 0.f, 0.f, 0.f);
}

// One wave per edge; 32 lanes x float4 = one 128-float row.
// Loop over batches reuses the src/dst fetch; 4 global_atomic_add_f32 per lane.
__global__ void edge_scatter(const float* __restrict__ emb,
                             const int* __restrict__ edge,
                             float* nb) {
  int gid  = blockIdx.x * blockDim.x + threadIdx.x;
  int e    = gid >> 5;
  int lane = gid & 31;
  if (e >= E) return;
  int src = edge[e];        // edge_index[0][e]
  int dst = edge[E + e];    // edge_index[1][e]
  const float* s = emb + (size_t)src * H + lane * 4;
  float*       d = nb  + (size_t)dst * H + lane * 4;
  const size_t bstride = (size_t)N * H;
  for (int b = 0; b < B; ++b) {
    float4 v = *(const float4*)(s + b * bstride);
    float* dp = d + b * bstride;
    atomicAdd(dp + 0, v.x);
    atomicAdd(dp + 1, v.y);
    atomicAdd(dp + 2, v.z);
    atomicAdd(dp + 3, v.w);
  }
}

// One wave per 16-node tile. ctx = [neighbor(128) | emb(128)] staged in LDS,
// GEMM1 (K=256) -> relu+bias -> LDS -> GEMM2 (K=128) -> bias+mask -> store.
// WMMA fragment layouts per CDNA5 ISA 7.12.2 (32-bit A 16x4 / B 4x16 / C 16x16).
__global__ void __launch_bounds__(WAVES * 32)
mlp_wmma(const float* nb,                    // neighbor accumulator (== d_out)
         const float* __restrict__ emb,
         const int*   __restrict__ mmask,
         const float* __restrict__ W1, const float* __restrict__ b1,
         const float* __restrict__ W2, const float* __restrict__ b2,
         float* out) {
  __shared__ float lds[WAVES * WSZ];
  const int lane = threadIdx.x & 31;
  const int wv   = threadIdx.x >> 5;
  const int t    = blockIdx.x * WAVES + wv;      // grid sized exactly: no divergence
  const int bb   = t / TPB;
  const int n0   = (t - bb * TPB) * 16;

  float* ctx  = lds + wv * WSZ;
  float* hbuf = ctx + 16 * CSTR;

  const size_t base = ((size_t)bb * N + n0) * H;

  // Cooperative per-wave load: neighbor -> ctx[:,0:128], emb -> ctx[:,128:256]
  for (int r = 0; r < 16; ++r) {
    const size_t off = base + (size_t)r * H + lane * 4;
    *(float4*)(ctx + r * CSTR + lane * 4)       = *(const float4*)(nb  + off);
    *(float4*)(ctx + r * CSTR + 128 + lane * 4) = *(const float4*)(emb + off);
  }

  const int hi   = lane >> 4;   // half-wave select
  const int lo   = lane & 15;
  const int koff = hi * 2;      // A/B K split: V0 -> K=koff, V1 -> K=koff+1

  v8f acc[8] = {};
  const v8f vzero = {};

  // GEMM1: (16x256) x W1(256x128), K-step 4 per WMMA
  for (int k = 0; k < 2 * H; k += 4) {
    v2f a = *(const v2f*)(ctx + lo * CSTR + k + koff);    // ds_load_b64, conflict-free
    const float* wrow = W1 + (size_t)(k + koff) * H;
    for (int nt = 0; nt < 8; ++nt) {
      int col = nt * 16 + lo;
      v2f bf;
      bf.x = wrow[col];
      bf.y = wrow[H + col];
      acc[nt] = __builtin_amdgcn_wmma_f32_16x16x4_f32(
          false, a, false, bf, (short)0, acc[nt], false, false);
    }
  }

  // bias + relu; transpose C-layout -> row-major hidden tile in LDS
  for (int nt = 0; nt < 8; ++nt) {
    int col = nt * 16 + lo;
    float bias = b1[col];
    for (int r = 0; r < 8; ++r) {
      float v = acc[nt][r] + bias;
      hbuf[(r + 8 * hi) * HSTR + col] = v > 0.f ? v : 0.f;
    }
    acc[nt] = vzero;
  }

  // GEMM2: (16x128) x W2(128x128)
  for (int k = 0; k < H; k += 4) {
    v2f a = *(const v2f*)(hbuf + lo * HSTR + k + koff);
    const float* wrow = W2 + (size_t)(k + koff) * H;
    for (int nt = 0; nt < 8; ++nt) {
      int col = nt * 16 + lo;
      v2f bf;
      bf.x = wrow[col];
      bf.y = wrow[H + col];
      acc[nt] = __builtin_amdgcn_wmma_f32_16x16x4_f32(
          false, a, false, bf, (short)0, acc[nt], false, false);
    }
  }

  // bias2 + missing-mask select + store (in place over the neighbor buffer)
  int mk[8];
  for (int r = 0; r < 8; ++r)
    mk[r] = mmask[bb * N + n0 + r + 8 * hi];
  for (int nt = 0; nt < 8; ++nt) {
    int col = nt * 16 + lo;
    float bias = b2[col];
    for (int r = 0; r < 8; ++r) {
      int row  = r + 8 * hi;
      float v  = acc[nt][r] + bias;
      float ev = ctx[row * CSTR + 128 + col];
      out[base + (size_t)row * H + col] = mk[r] ? v : ev;
    }
  }
}

extern "C" void kernel_launch(void* const* d_in, const int* in_sizes, int n_in,
                              void* d_out, int out_size, void* d_ws, size_t ws_size,
                              hipStream_t stream) {
  const float* emb   = (const float*)d_in[0];
  const int*   mmask = (const int*)  d_in[1];
  const int*   edge  = (const int*)  d_in[2];
  const float* W1    = (const float*)d_in[3];
  const float* b1    = (const float*)d_in[4];
  const float* W2    = (const float*)d_in[5];
  const float* b2    = (const float*)d_in[6];
  float* out = (float*)d_out;

  // Phase 0: zero the neighbor accumulator (d_out doubles as scratch; harness
  // poisons it, so re-zero every call).
  const int n4 = (B * N * H) / 4;  // 5,120,000 float4s
  zero_f4<<<(n4 + 255) / 256, 256, 0, stream>>>((float4*)out, n4);

  // Phase 1: segment-sum via f32 atomics (one wave per edge).
  const long threads = (long)E * 32;
  edge_scatter<<<(int)((threads + 255) / 256), 256, 0, stream>>>(emb, edge, out);

  // Phase 2: WMMA MLP + mask blend, in place over d_out.
  mlp_wmma<<<TILES / WAVES, WAVES * 32, 0, stream>>>(out, emb, mmask,
                                                     W1, b1, W2, b2, out);
}